// ASTGPool_57475252355503
// MI455X (gfx1250) — compile-verified
//
#include <hip/hip_runtime.h>
#include <stdint.h>

// Problem constants (match reference)
#define G_      64
#define N_      1024
#define E_      1048576
#define F_      256
#define K_      512
#define NNODES  (G_ * N_)   // 65536
#define NROWS   (G_ * K_)   // 32768

// d_out layout (float element offsets), tuple concatenated flat:
// x_filtered [G*K, F], remapped_edge_index [2, E], batch [G*K],
// perm [G*K], score [G*K], edge_mask [E]
#define OX   0LL
#define OEI  8388608LL              // G*K*F
#define OB   (OEI + 2LL * E_)       // 10485760
#define OP   (OB + (long long)NROWS)
#define OS   (OP + (long long)NROWS)
#define OM   (OS + (long long)NROWS)

// ---------------------------------------------------------------------------
// CDNA5 async global<->LDS DMA helpers (gfx1250; ASYNCcnt-tracked)
// ---------------------------------------------------------------------------
template <int OFF>
__device__ __forceinline__ void async_load_b128(uint32_t lds_addr, uint64_t gaddr) {
    asm volatile("global_load_async_to_lds_b128 %0, %1, off offset:%2"
                 :: "v"(lds_addr), "v"(gaddr), "i"(OFF) : "memory");
}

template <int OFF>
__device__ __forceinline__ void async_store_b128(uint64_t gaddr, uint32_t lds_addr) {
    asm volatile("global_store_async_from_lds_b128 %0, %1, off offset:%2"
                 :: "v"(gaddr), "v"(lds_addr), "i"(OFF) : "memory");
}

__device__ __forceinline__ void wait_async0() {
    asm volatile("s_wait_asynccnt 0x0" ::: "memory");
}

// ---------------------------------------------------------------------------
// Kernel 1a: zero the degree array in workspace
// ---------------------------------------------------------------------------
__global__ void k_zero_deg(unsigned* __restrict__ deg) {
    int i = blockIdx.x * blockDim.x + threadIdx.x;
    if (i < NNODES) deg[i] = 0u;
}

// ---------------------------------------------------------------------------
// Kernel 1b: out-degree scatter-add over edge sources
// ---------------------------------------------------------------------------
__global__ void k_degree(const int* __restrict__ src, unsigned* __restrict__ deg) {
    int e = blockIdx.x * blockDim.x + threadIdx.x;
    if (e < E_) atomicAdd(&deg[src[e]], 1u);
}

// ---------------------------------------------------------------------------
// Kernel 2: per-graph stable top-K via exact-rank computation.
// One block (1024 threads) per graph. key = (deg<<10) | (1023 - i) makes all
// keys distinct; rank(i) = #{j : key[j] > key[i]} is this node's position in
// the stable descending sort — identical ordering to jax.lax.top_k.
// ---------------------------------------------------------------------------
__global__ void k_topk(const unsigned* __restrict__ deg,
                       int* __restrict__ node_map,
                       int* __restrict__ perm,
                       float* __restrict__ out) {
    __shared__ unsigned keys[N_];
    const int g = blockIdx.x;
    const int i = threadIdx.x;

    const unsigned d = deg[g * N_ + i];
    const unsigned mykey = (d << 10) | (unsigned)(N_ - 1 - i);
    keys[i] = mykey;
    __syncthreads();

    int rank = 0;
    #pragma unroll 8
    for (int j = 0; j < N_; ++j) rank += (keys[j] > mykey) ? 1 : 0;

    const int gid = g * N_ + i;
    if (rank < K_) {
        const int p = g * K_ + rank;   // position in perm (sorted order)
        perm[p]      = gid;
        node_map[gid] = p;
        out[OP + p] = (float)gid;      // perm
        out[OS + p] = (float)d;        // score_filtered
        out[OB + p] = (float)g;        // batch_filtered
    } else {
        node_map[gid] = -1;
    }
}

// ---------------------------------------------------------------------------
// Kernel 3: edge mask + id remap.  node_map is 256KB -> L2-resident gathers.
// ---------------------------------------------------------------------------
__global__ void k_edges(const int* __restrict__ ei,
                        const int* __restrict__ node_map,
                        float* __restrict__ out) {
    int e = blockIdx.x * blockDim.x + threadIdx.x;
    if (e >= E_) return;
    const int s = ei[e];
    const int t = ei[E_ + e];
    const int ms = node_map[s];
    const int mt = node_map[t];
    const bool m = (ms >= 0) && (mt >= 0);
    out[OEI + e]        = m ? (float)ms : -1.0f;
    out[OEI + E_ + e]   = m ? (float)mt : -1.0f;
    out[OM + e]         = m ? 1.0f : 0.0f;
}

// ---------------------------------------------------------------------------
// Kernel 4: x[perm] row gather via CDNA5 async global->LDS->global DMA.
// Each wave copies 1KB rows (2 x b128-per-lane = 512B per issue), double-
// buffered in LDS: stores of row r overlap loads of row r+stride, with one
// s_wait_asynccnt 0 drain per row.  No VGPR data round-trip.
// ---------------------------------------------------------------------------
__global__ void k_gather(const float* __restrict__ x,
                         const int* __restrict__ perm,
                         float* __restrict__ outx) {
    __shared__ __align__(16) unsigned char smem[8 * 2048]; // 8 waves * 2 * 1KB
    const unsigned lane = threadIdx.x & 31u;
    const unsigned wib  = threadIdx.x >> 5;                // wave in block
    const unsigned gwid = blockIdx.x * (blockDim.x >> 5) + wib;
    const unsigned nw   = gridDim.x * (blockDim.x >> 5);

    // Workgroup-relative LDS byte addresses (low 32 bits of generic pointer)
    const uint32_t lds0 = (uint32_t)(uint64_t)(uintptr_t)(&smem[wib * 2048]) + lane * 16u;
    const uint32_t lds1 = lds0 + 1024u;

    int r = (int)gwid;
    if (r < NROWS) {
        const int p = perm[r];
        const uint64_t ga = (uint64_t)(uintptr_t)x + ((uint64_t)p << 10) + (uint64_t)lane * 16u;
        async_load_b128<0>(lds0, ga);
        async_load_b128<512>(lds0, ga);
    }
    unsigned buf = 0u;
    for (; r < NROWS; r += (int)nw) {
        wait_async0(); // row r loaded; prior stores drained (safe to reuse LDS)
        const uint32_t lcur = buf ? lds1 : lds0;
        const uint64_t wa = (uint64_t)(uintptr_t)outx + ((uint64_t)r << 10) + (uint64_t)lane * 16u;
        async_store_b128<0>(wa, lcur);
        async_store_b128<512>(wa, lcur);
        const int rn = r + (int)nw;
        if (rn < NROWS) {
            const int p = perm[rn];
            const uint64_t ga = (uint64_t)(uintptr_t)x + ((uint64_t)p << 10) + (uint64_t)lane * 16u;
            const uint32_t lnext = buf ? lds0 : lds1;
            async_load_b128<0>(lnext, ga);
            async_load_b128<512>(lnext, ga);
        }
        buf ^= 1u;
    }
    // implicit S_WAIT_IDLE at s_endpgm drains outstanding async stores
}

// ---------------------------------------------------------------------------
extern "C" void kernel_launch(void* const* d_in, const int* in_sizes, int n_in,
                              void* d_out, int out_size, void* d_ws, size_t ws_size,
                              hipStream_t stream) {
    const float* x  = (const float*)d_in[0];   // [G*N, F] f32
    const int*   ei = (const int*)d_in[1];     // [2, E] i32
    // d_in[2] (batch) unused: batch_filtered == graph id, computed directly.
    float* out = (float*)d_out;

    // Workspace layout: degree (64K u32) | node_map (64K i32) | perm (32K i32)
    unsigned* deg  = (unsigned*)d_ws;
    int* node_map  = (int*)((char*)d_ws + (size_t)NNODES * 4);
    int* perm      = (int*)((char*)d_ws + (size_t)NNODES * 8);

    k_zero_deg<<<NNODES / 256, 256, 0, stream>>>(deg);
    k_degree  <<<E_ / 256, 256, 0, stream>>>(ei, deg);
    k_topk    <<<G_, N_, 0, stream>>>(deg, node_map, perm, out);
    k_edges   <<<E_ / 256, 256, 0, stream>>>(ei, node_map, out);
    k_gather  <<<512, 256, 0, stream>>>(x, perm, out + OX);
}